// GCNNet_38508676776214
// MI455X (gfx1250) — compile-verified
//
#include <hip/hip_runtime.h>
#include <hip/hip_bf16.h>

typedef float v2f __attribute__((ext_vector_type(2)));
typedef float v8f __attribute__((ext_vector_type(8)));

#define N_NODES 100000
#define N_EDGES 1600000
#define NGRAPH  64
#define FDIM    128
#define FOUT    32
#define BN_EPS  1e-5f

// ---------------- utility ----------------
__global__ void fill_f32v(float4* __restrict__ p, float v, long n4) {
    long i = (long)blockIdx.x * blockDim.x + threadIdx.x;
    if (i < n4) p[i] = make_float4(v, v, v, v);
}

// transpose W[K,Fout] -> Wt[Fout,K] so B fragments load contiguously
__global__ void transpose_w(const float* __restrict__ in, float* __restrict__ out,
                            int K, int Fo) {
    int i = blockIdx.x * blockDim.x + threadIdx.x;
    if (i < K * Fo) {
        int k = i / Fo, c = i % Fo;
        out[c * K + k] = in[i];
    }
}

// ---------------- degree / norm ----------------
__global__ void deg_accum(const int* __restrict__ dst, float* __restrict__ deg, int ne) {
    int i = blockIdx.x * blockDim.x + threadIdx.x;
    if (i < ne) unsafeAtomicAdd(&deg[dst[i]], 1.0f);
}

__global__ void dinv_kernel(float* __restrict__ deg, int n) {
    int i = blockIdx.x * blockDim.x + threadIdx.x;
    if (i < n) {
        float d = deg[i];
        deg[i] = (d > 0.0f) ? rsqrtf(d) : 0.0f;
    }
}

// ---------------- dense GEMM via V_WMMA_F32_16X16X4_F32 ----------------
// One wave computes a full 16-row x FoutT strip of H = X[nrows,128] @ W[128,FoutT].
// NT = FoutT/16 accumulators; A fragment loaded once per k-step and reused.
// A-frag (16x4 f32): row = lane&15, khalf = lane>>4, VGPR0/1 = K = 2*kh, 2*kh+1.
// B-frag (4x16 f32): col = lane&15, same khalf split; Wt[Fout,K] gives contiguous b64.
// C/D (16x16 f32, 8 VGPRs): m = kh*8 + j, n = lane&15.
template <int FoutT>
__global__ __launch_bounds__(256) void gemm_wmma_f32(
    const float* __restrict__ X, const float* __restrict__ Wt,
    float* __restrict__ H, int nrows) {
    constexpr int NT = FoutT / 16;
    const int wave = blockIdx.x * (blockDim.x >> 5) + (threadIdx.x >> 5);
    if (wave >= nrows / 16) return;         // wave-uniform: EXEC all-1s inside

    const int tm   = wave;
    const int lane = threadIdx.x & 31;
    const int r    = lane & 15;             // A row-in-tile / B col-in-tile
    const int kh   = lane >> 4;             // K half selector

    const float* xrow = X + (long)(tm * 16 + r) * FDIM + 2 * kh;
    const float* wrow = Wt + (long)r * FDIM + 2 * kh;   // + t*16*FDIM per n-tile

    v8f c[NT];
#pragma unroll
    for (int t = 0; t < NT; ++t) c[t] = (v8f){0.f, 0.f, 0.f, 0.f, 0.f, 0.f, 0.f, 0.f};

#pragma unroll
    for (int k0 = 0; k0 < FDIM; k0 += 4) {
        v2f a = *(const v2f*)(xrow + k0);
#pragma unroll
        for (int t = 0; t < NT; ++t) {
            v2f b = *(const v2f*)(wrow + (long)t * 16 * FDIM + k0);
            c[t] = __builtin_amdgcn_wmma_f32_16x16x4_f32(
                       false, a, false, b, (short)0, c[t], false, false);
        }
    }

#pragma unroll
    for (int t = 0; t < NT; ++t) {
        float* hout = H + (long)(tm * 16 + kh * 8) * FoutT + t * 16 + r;
#pragma unroll
        for (int j = 0; j < 8; ++j) hout[(long)j * FoutT] = c[t][j];
    }
}

// ---------------- edge scatter:  out[dst] += dinv[src]*dinv[dst] * h[src] ----------------
// float4 gathers; F=128 -> 1 edge per wave, F=32 -> 4 edges per wave.
template <int F>
__global__ void scatter_edges(const float* __restrict__ h,
                              const int* __restrict__ src, const int* __restrict__ dst,
                              const float* __restrict__ dinv,
                              float* __restrict__ out, int ne) {
    constexpr int LPE = F / 4;      // lanes per edge
    constexpr int EPW = 32 / LPE;   // edges per wave
    int wave = blockIdx.x * (blockDim.x >> 5) + (threadIdx.x >> 5);
    int lane = threadIdx.x & 31;
    int e = wave * EPW + lane / LPE;
    if (e >= ne) return;
    int fb = (lane % LPE) * 4;
    int s = src[e], d = dst[e];
    float w = dinv[s] * dinv[d];
    const float4 hv = *(const float4*)(h + (long)s * F + fb);
    float* o = out + (long)d * F + fb;
    unsafeAtomicAdd(o + 0, w * hv.x);
    unsafeAtomicAdd(o + 1, w * hv.y);
    unsafeAtomicAdd(o + 2, w * hv.z);
    unsafeAtomicAdd(o + 3, w * hv.w);
}

// self-loop contribution + bias:  out += dinv^2 * h + b   (float4)
template <int F>
__global__ void finalize_selfloop_bias(float* __restrict__ out, const float* __restrict__ h,
                                       const float* __restrict__ dinv,
                                       const float* __restrict__ bias, int n) {
    constexpr int F4 = F / 4;
    long i4 = (long)blockIdx.x * blockDim.x + threadIdx.x;
    if (i4 >= (long)n * F4) return;
    int row = (int)(i4 / F4);
    int fb  = (int)(i4 % F4) * 4;
    float di = dinv[row];
    float w = di * di;
    float4 o  = ((float4*)out)[i4];
    float4 hv = ((const float4*)h)[i4];
    float4 bv = *(const float4*)(bias + fb);
    o.x += w * hv.x + bv.x;
    o.y += w * hv.y + bv.y;
    o.z += w * hv.z + bv.z;
    o.w += w * hv.w + bv.w;
    ((float4*)out)[i4] = o;
}

// ---------------- BatchNorm (training stats) + ReLU ----------------
__global__ void bn_stats(const float* __restrict__ x, float* __restrict__ sums,
                         float* __restrict__ sqs, int n) {
    int lane = threadIdx.x & 31;
    int fb = lane * 4;                                   // 32 lanes x 4 = 128 features
    int rs = blockIdx.x * (blockDim.x >> 5) + (threadIdx.x >> 5);
    int stride = gridDim.x * (blockDim.x >> 5);
    float4 s  = make_float4(0.f, 0.f, 0.f, 0.f);
    float4 s2 = make_float4(0.f, 0.f, 0.f, 0.f);
    for (int r = rs; r < n; r += stride) {
        float4 v = *(const float4*)(x + (long)r * FDIM + fb);
        s.x += v.x; s.y += v.y; s.z += v.z; s.w += v.w;
        s2.x += v.x * v.x; s2.y += v.y * v.y; s2.z += v.z * v.z; s2.w += v.w * v.w;
    }
    unsafeAtomicAdd(&sums[fb + 0], s.x);
    unsafeAtomicAdd(&sums[fb + 1], s.y);
    unsafeAtomicAdd(&sums[fb + 2], s.z);
    unsafeAtomicAdd(&sums[fb + 3], s.w);
    unsafeAtomicAdd(&sqs[fb + 0], s2.x);
    unsafeAtomicAdd(&sqs[fb + 1], s2.y);
    unsafeAtomicAdd(&sqs[fb + 2], s2.z);
    unsafeAtomicAdd(&sqs[fb + 3], s2.w);
}

__global__ void bn_finalize(const float* __restrict__ sums, const float* __restrict__ sqs,
                            const float* __restrict__ gamma, const float* __restrict__ beta,
                            float* __restrict__ scale, float* __restrict__ shift, int n) {
    int f = threadIdx.x;
    if (f < FDIM) {
        float invn = 1.0f / (float)n;
        float mean = sums[f] * invn;
        float var  = sqs[f] * invn - mean * mean;
        float sc   = gamma[f] * rsqrtf(var + BN_EPS);
        scale[f] = sc;
        shift[f] = beta[f] - mean * sc;
    }
}

__global__ void bn_apply_relu(float* __restrict__ x, const float* __restrict__ scale,
                              const float* __restrict__ shift, long total4) {
    long i4 = (long)blockIdx.x * blockDim.x + threadIdx.x;
    if (i4 >= total4) return;
    int fb = (int)(i4 & (FDIM / 4 - 1)) * 4;
    float4 v  = ((float4*)x)[i4];
    float4 sc = *(const float4*)(scale + fb);
    float4 sh = *(const float4*)(shift + fb);
    v.x = fmaxf(v.x * sc.x + sh.x, 0.f);
    v.y = fmaxf(v.y * sc.y + sh.y, 0.f);
    v.z = fmaxf(v.z * sc.z + sh.z, 0.f);
    v.w = fmaxf(v.w * sc.w + sh.w, 0.f);
    ((float4*)x)[i4] = v;
}

// ---------------- global mean pool + sigmoid ----------------
__global__ void pool_count(const int* __restrict__ batch, float* __restrict__ cnt, int n) {
    int i = blockIdx.x * blockDim.x + threadIdx.x;
    if (i < n) unsafeAtomicAdd(&cnt[batch[i]], 1.0f);
}

__global__ void pool_sum(const float* __restrict__ h, const int* __restrict__ batch,
                         float* __restrict__ pooled, int n) {
    long i = (long)blockIdx.x * blockDim.x + threadIdx.x;      // over n * (FOUT/4)
    if (i >= (long)n * (FOUT / 4)) return;
    int row = (int)(i >> 3);            // FOUT/4 == 8
    int fb  = (int)(i & 7) * 4;
    float4 v = *(const float4*)(h + (long)row * FOUT + fb);
    float* p = pooled + batch[row] * FOUT + fb;
    unsafeAtomicAdd(p + 0, v.x);
    unsafeAtomicAdd(p + 1, v.y);
    unsafeAtomicAdd(p + 2, v.z);
    unsafeAtomicAdd(p + 3, v.w);
}

__global__ void pool_final(const float* __restrict__ pooled, const float* __restrict__ cnt,
                           float* __restrict__ out) {
    int i = blockIdx.x * blockDim.x + threadIdx.x;
    if (i >= NGRAPH * FOUT) return;
    int g = i / FOUT;
    float c = cnt[g];
    float m = pooled[i] / (c > 1.0f ? c : 1.0f);
    out[i] = 1.0f / (1.0f + expf(-m));
}

// ---------------- driver ----------------
extern "C" void kernel_launch(void* const* d_in, const int* in_sizes, int n_in,
                              void* d_out, int out_size, void* d_ws, size_t ws_size,
                              hipStream_t stream) {
    const float* x      = (const float*)d_in[0];
    const int*   ei     = (const int*)d_in[1];   // [2,E]: src = ei, dst = ei+E
    const int*   batch  = (const int*)d_in[2];
    const float* W1     = (const float*)d_in[3];
    const float* b1     = (const float*)d_in[4];
    const float* W2     = (const float*)d_in[5];
    const float* b2     = (const float*)d_in[6];
    const float* W3     = (const float*)d_in[7];
    const float* b3     = (const float*)d_in[8];
    const float* gamma1 = (const float*)d_in[9];
    const float* beta1  = (const float*)d_in[10];
    const float* gamma2 = (const float*)d_in[11];
    const float* beta2  = (const float*)d_in[12];
    const int* src = ei;
    const int* dst = ei + N_EDGES;

    // workspace carve-up
    float* G      = (float*)d_ws;                         // [N,128] gemm out
    float* S      = G + (size_t)N_NODES * FDIM;           // [N,128] scatter/activation
    float* dinv   = S + (size_t)N_NODES * FDIM;           // [N]
    float* bnsum  = dinv + N_NODES;                       // [128]
    float* bnsq   = bnsum + FDIM;                         // [128]
    float* bnsc   = bnsq + FDIM;                          // [128]
    float* bnsh   = bnsc + FDIM;                          // [128]
    float* pooled = bnsh + FDIM;                          // [64*32]
    float* pcnt   = pooled + NGRAPH * FOUT;               // [64]
    float* Wt1    = pcnt + NGRAPH;                        // [128,128] transposed
    float* Wt2    = Wt1 + FDIM * FDIM;                    // [128,128]
    float* Wt3    = Wt2 + FDIM * FDIM;                    // [32,128]

    const long NF     = (long)N_NODES * FDIM;
    const long NF3    = (long)N_NODES * FOUT;
    const int  blkNF4  = (int)((NF / 4 + 255) / 256);
    const int  blkNF34 = (int)((NF3 / 4 + 255) / 256);
    const int  blkN    = (N_NODES + 255) / 256;
    const int  blkE    = (N_EDGES + 255) / 256;
    const int  gemmBlk = ((N_NODES / 16) + 7) / 8;        // 1 wave per 16-row strip

    // ---- weight transposes (cheap, deterministic per call) ----
    transpose_w<<<(FDIM * FDIM + 255) / 256, 256, 0, stream>>>(W1, Wt1, FDIM, FDIM);
    transpose_w<<<(FDIM * FDIM + 255) / 256, 256, 0, stream>>>(W2, Wt2, FDIM, FDIM);
    transpose_w<<<(FDIM * FOUT + 255) / 256, 256, 0, stream>>>(W3, Wt3, FDIM, FOUT);

    // ---- symmetric GCN normalization ----
    fill_f32v<<<(N_NODES / 4 + 255) / 256, 256, 0, stream>>>((float4*)dinv, 1.0f, N_NODES / 4);
    deg_accum<<<blkE, 256, 0, stream>>>(dst, dinv, N_EDGES);
    dinv_kernel<<<blkN, 256, 0, stream>>>(dinv, N_NODES);

    // ---- layer 1: GCNConv(128->128) + BN + ReLU ----
    gemm_wmma_f32<FDIM><<<gemmBlk, 256, 0, stream>>>(x, Wt1, G, N_NODES);
    fill_f32v<<<blkNF4, 256, 0, stream>>>((float4*)S, 0.0f, NF / 4);
    scatter_edges<FDIM><<<N_EDGES / 8, 256, 0, stream>>>(G, src, dst, dinv, S, N_EDGES);
    finalize_selfloop_bias<FDIM><<<blkNF4, 256, 0, stream>>>(S, G, dinv, b1, N_NODES);
    fill_f32v<<<1, 64, 0, stream>>>((float4*)bnsum, 0.0f, 2 * FDIM / 4);
    bn_stats<<<512, 256, 0, stream>>>(S, bnsum, bnsq, N_NODES);
    bn_finalize<<<1, FDIM, 0, stream>>>(bnsum, bnsq, gamma1, beta1, bnsc, bnsh, N_NODES);
    bn_apply_relu<<<blkNF4, 256, 0, stream>>>(S, bnsc, bnsh, NF / 4);

    // ---- layer 2: GCNConv(128->128) + BN + ReLU ----
    gemm_wmma_f32<FDIM><<<gemmBlk, 256, 0, stream>>>(S, Wt2, G, N_NODES);
    fill_f32v<<<blkNF4, 256, 0, stream>>>((float4*)S, 0.0f, NF / 4);
    scatter_edges<FDIM><<<N_EDGES / 8, 256, 0, stream>>>(G, src, dst, dinv, S, N_EDGES);
    finalize_selfloop_bias<FDIM><<<blkNF4, 256, 0, stream>>>(S, G, dinv, b2, N_NODES);
    fill_f32v<<<1, 64, 0, stream>>>((float4*)bnsum, 0.0f, 2 * FDIM / 4);
    bn_stats<<<512, 256, 0, stream>>>(S, bnsum, bnsq, N_NODES);
    bn_finalize<<<1, FDIM, 0, stream>>>(bnsum, bnsq, gamma2, beta2, bnsc, bnsh, N_NODES);
    bn_apply_relu<<<blkNF4, 256, 0, stream>>>(S, bnsc, bnsh, NF / 4);

    // ---- layer 3: GCNConv(128->32) ----
    gemm_wmma_f32<FOUT><<<gemmBlk, 256, 0, stream>>>(S, Wt3, G, N_NODES);  // G as [N,32]
    float* T3 = S;                                                         // reuse S front
    fill_f32v<<<blkNF34, 256, 0, stream>>>((float4*)T3, 0.0f, NF3 / 4);
    scatter_edges<FOUT><<<N_EDGES / 32, 256, 0, stream>>>(G, src, dst, dinv, T3, N_EDGES);
    finalize_selfloop_bias<FOUT><<<blkNF34, 256, 0, stream>>>(T3, G, dinv, b3, N_NODES);

    // ---- global mean pool + sigmoid ----
    fill_f32v<<<1, 256, 0, stream>>>((float4*)pooled, 0.0f, (NGRAPH * FOUT + NGRAPH) / 4);
    pool_count<<<blkN, 256, 0, stream>>>(batch, pcnt, N_NODES);
    pool_sum<<<(int)((NF3 / 4 + 255) / 256), 256, 0, stream>>>(T3, batch, pooled, N_NODES);
    pool_final<<<(NGRAPH * FOUT + 255) / 256, 256, 0, stream>>>(pooled, pcnt, (float*)d_out);
}